// Encoder_34557306863777
// MI455X (gfx1250) — compile-verified
//
#include <hip/hip_runtime.h>

typedef __attribute__((ext_vector_type(2))) float v2f;
typedef __attribute__((ext_vector_type(8))) float v8f;

#define DIM 64        // input feature dim
#define HID 128       // output dim (= 2*DIM, also K of the GEMM)
#define ROW_TILES_PER_BLOCK 16   // 16 tiles * 16 rows = 256 nodes per block

// ---------------------------------------------------------------------------
// Zero the accumulator workspace (neigh_sum [N,64] + deg [N]).
// ---------------------------------------------------------------------------
__global__ void zero_ws(float* __restrict__ ws, int n) {
  int i = blockIdx.x * blockDim.x + threadIdx.x;
  if (i < n) ws[i] = 0.0f;
}

// ---------------------------------------------------------------------------
// Undirected edge scatter: for edge (s,d) add feat[d] into sum[s], feat[s]
// into sum[d], and bump both degrees. 16 lanes per edge, 4 floats per lane.
// ---------------------------------------------------------------------------
__global__ void __launch_bounds__(256) edge_scatter(
    const float* __restrict__ feat,
    const int*   __restrict__ src_idx,
    const int*   __restrict__ dst_idx,
    float* __restrict__ nsum,
    float* __restrict__ deg,
    int E) {
  int gid = blockIdx.x * blockDim.x + threadIdx.x;
  int e = gid >> 4;
  int l = gid & 15;
  if (e >= E) return;
  int s = src_idx[e];
  int d = dst_idx[e];
  const float4 fs = ((const float4*)(feat + (size_t)s * DIM))[l];
  const float4 fd = ((const float4*)(feat + (size_t)d * DIM))[l];
  float* ps = nsum + (size_t)s * DIM + l * 4;
  float* pd = nsum + (size_t)d * DIM + l * 4;
  atomicAdd(ps + 0, fd.x); atomicAdd(ps + 1, fd.y);
  atomicAdd(ps + 2, fd.z); atomicAdd(ps + 3, fd.w);
  atomicAdd(pd + 0, fs.x); atomicAdd(pd + 1, fs.y);
  atomicAdd(pd + 2, fs.z); atomicAdd(pd + 3, fs.w);
  if (l == 0) {
    atomicAdd(deg + s, 1.0f);
    atomicAdd(deg + d, 1.0f);
  }
}

// ---------------------------------------------------------------------------
// Fused concat + Linear + ReLU via V_WMMA_F32_16X16X4_F32.
//   out[m][h] = relu( sum_k combined[m][k] * W[h][k] + b[h] )
//   combined[m][k] = (k < 64) ? feat[m][k] : nsum[m][k-64] / max(deg[m],1)
// Block = 256 threads = 8 waves; wave w owns output columns [16w, 16w+16).
// Each wave preloads its B strip (W^T columns) into 64 VGPRs, then sweeps
// 16 row tiles. Per tile: batch-load all 32 A fragments (one load clause,
// one wait), then issue 32 back-to-back WMMAs; invd scaling co-executes on
// the VALU. A/B fragment layout for 16x16x4 f32 (ISA 7.12.2):
//   lanes 0-15:  VGPR0 -> K=4s+0, VGPR1 -> K=4s+1   (M/N = lane)
//   lanes 16-31: VGPR0 -> K=4s+2, VGPR1 -> K=4s+3   (M/N = lane-16)
// C/D layout: VGPR v -> row tileM + v + 8*(lane>=16), col = tileN + (lane&15)
// ---------------------------------------------------------------------------
__global__ void __launch_bounds__(256) sage_gemm(
    const float* __restrict__ feat,   // [N, 64]
    const float* __restrict__ nsum,   // [N, 64]
    const float* __restrict__ deg,    // [N]
    const float* __restrict__ W,      // [128, 128] row-major (H x 2D)
    const float* __restrict__ bias,   // [128]
    float* __restrict__ out,          // [N, 128]
    int nNodes) {
  const int wave = threadIdx.x >> 5;   // 0..7 -> column tile
  const int lane = threadIdx.x & 31;
  const int lh   = lane & 15;
  const int sel  = lane >> 4;          // 0: K offset {0,1}; 1: K offset {2,3}
  const int colBase = wave * 16;

  // Preload B strip: Bfrag[s] covers K = 4s..4s+3 for column colBase+lh.
  v2f Bfrag[32];
  {
    const float* wrow = W + (size_t)(colBase + lh) * HID + 2 * sel;
    #pragma unroll
    for (int s = 0; s < 32; ++s) {
      Bfrag[s] = *(const v2f*)(wrow + 4 * s);
    }
  }
  const float bcol = bias[colBase + lh];

  const int rowTile0 = blockIdx.x * ROW_TILES_PER_BLOCK;
  for (int rt = 0; rt < ROW_TILES_PER_BLOCK; ++rt) {
    const int mBase = (rowTile0 + rt) * 16;     // uniform across wave
    if (mBase >= nNodes) break;                 // uniform branch: EXEC stays full
    const int m  = mBase + lh;
    const int mc = m < nNodes ? m : (nNodes - 1);   // clamp loads; stores guarded
    const float* frow = feat + (size_t)mc * DIM + 2 * sel;
    const float* srow = nsum + (size_t)mc * DIM + 2 * sel;

    // Batch all A-fragment loads so they form one clause with a single wait.
    v2f Afeat[16];
    v2f Asum[16];
    #pragma unroll
    for (int s = 0; s < 16; ++s) Afeat[s] = *(const v2f*)(frow + 4 * s);
    #pragma unroll
    for (int s = 0; s < 16; ++s) Asum[s]  = *(const v2f*)(srow + 4 * s);
    // Fast reciprocal: deg is an exact small integer in fp32; v_rcp_f32 (1ulp)
    // is plenty and replaces the ~10-op IEEE divide chain.
    const float invd = __builtin_amdgcn_rcpf(fmaxf(deg[mc], 1.0f));

    v8f acc = {0.f, 0.f, 0.f, 0.f, 0.f, 0.f, 0.f, 0.f};

    #pragma unroll
    for (int s = 0; s < 16; ++s) {   // K = 0..63 : self features
      acc = __builtin_amdgcn_wmma_f32_16x16x4_f32(
          false, Afeat[s], false, Bfrag[s], (short)0, acc, false, false);
    }
    #pragma unroll
    for (int s = 0; s < 16; ++s) {   // K = 64..127 : neighbor mean
      v2f a;
      a.x = Asum[s].x * invd;
      a.y = Asum[s].y * invd;
      acc = __builtin_amdgcn_wmma_f32_16x16x4_f32(
          false, a, false, Bfrag[16 + s], (short)0, acc, false, false);
    }

    #pragma unroll
    for (int v = 0; v < 8; ++v) {
      const int r = mBase + v + 8 * sel;
      if (r < nNodes) {
        const float x = acc[v] + bcol;
        // Output is write-once, never re-read: non-temporal store keeps L2
        // free for feat/nsum reuse.
        __builtin_nontemporal_store(fmaxf(x, 0.0f),
                                    out + (size_t)r * HID + colBase + lh);
      }
    }
  }
}

// ---------------------------------------------------------------------------
// Launcher. Inputs: [0]=nodes(int N), [1]=features(f32 N*64),
// [2]=edge_index(int 2*E), [3]=W(f32 128*128), [4]=b(f32 128).
// Output: f32 N*128. Workspace: nsum (N*64 f32) + deg (N f32) = ~26 MB.
// ---------------------------------------------------------------------------
extern "C" void kernel_launch(void* const* d_in, const int* in_sizes, int n_in,
                              void* d_out, int out_size, void* d_ws, size_t ws_size,
                              hipStream_t stream) {
  const float* feat = (const float*)d_in[1];
  const int*   ei   = (const int*)d_in[2];
  const float* W    = (const float*)d_in[3];
  const float* b    = (const float*)d_in[4];
  float*       outp = (float*)d_out;

  const int nNodes = in_sizes[0];
  const int E      = in_sizes[2] / 2;

  float* nsum = (float*)d_ws;
  float* degw = nsum + (size_t)nNodes * DIM;

  // 1) zero accumulators
  const int zn = nNodes * (DIM + 1);
  zero_ws<<<(zn + 255) / 256, 256, 0, stream>>>(nsum, zn);

  // 2) edge scatter (16 lanes per edge)
  const long long sthreads = (long long)E * 16;
  const int sblocks = (int)((sthreads + 255) / 256);
  edge_scatter<<<sblocks, 256, 0, stream>>>(feat, ei, ei + E, nsum, degw, E);

  // 3) fused concat + GEMM + bias + relu
  const int rowsPerBlock = 16 * ROW_TILES_PER_BLOCK;  // 256 nodes
  const int gblocks = (nNodes + rowsPerBlock - 1) / rowsPerBlock;
  sage_gemm<<<gblocks, 256, 0, stream>>>(feat, nsum, degw, W, b, outp, nNodes);
}